// CenterExtractor_22539988370119
// MI455X (gfx1250) — compile-verified
//
#include <hip/hip_runtime.h>

// CenterExtractor zones branch for MI455X (gfx1250, wave32).
// Segmented max over 256 zones per batch image, then equality mask + count.
// Memory-bound: ~200 MB of HBM traffic -> ~9 us roofline at 23.3 TB/s.

#define NUM_ZONES 256
#define TPB 256      // 8 wave32 waves per block
#define VEC 4        // float4 / int4 (B128) per thread

typedef float  __attribute__((ext_vector_type(4))) f32x4;
typedef int    __attribute__((ext_vector_type(4))) i32x4;

// Monotonic order-preserving map f32 -> u32.
// f >= 0 : bits | 0x80000000 ; f < 0 : ~bits.
// No real input maps to 0 (only bit-pattern 0xFFFFFFFF = -NaN does),
// so 0 is a safe "empty zone" sentinel that matches nothing -- mirrors
// the reference's seg_max = -inf for empty zones.
__device__ __forceinline__ unsigned int fmap(float f) {
    unsigned int u = __float_as_uint(f);
    return (u & 0x80000000u) ? ~u : (u | 0x80000000u);
}

// ---------------------------------------------------------------------------
// Kernel 0: zero workspace (seg-max table + counter). Must run every launch:
// the harness poisons d_ws once and never re-poisons between graph replays.
// ---------------------------------------------------------------------------
__global__ void ce_init_ws(unsigned int* __restrict__ ws, int n) {
    int i = blockIdx.x * blockDim.x + threadIdx.x;
    if (i < n) ws[i] = 0u;
}

// ---------------------------------------------------------------------------
// Kernel 1: per-batch segmented max.
// blockIdx.y = batch. LDS ds_max_u32 atomics into a 256-entry table,
// then one global_atomic_max_u32 per non-empty zone per block.
// ---------------------------------------------------------------------------
__global__ void ce_seg_max(const float* __restrict__ x,
                           const int*   __restrict__ z,
                           unsigned int* __restrict__ gmax,
                           int n_per_batch) {
    __shared__ unsigned int smax[NUM_ZONES];
    const int tid = threadIdx.x;
    smax[tid] = 0u;
    __syncthreads();

    const size_t bbase  = (size_t)blockIdx.y * (size_t)n_per_batch;
    const int    stride = (int)gridDim.x * TPB * VEC;

    for (int off = ((int)blockIdx.x * TPB + tid) * VEC; off < n_per_batch; off += stride) {
        const f32x4 xv = *(const f32x4*)(x + bbase + off);
        const i32x4 zv = *(const i32x4*)(z + bbase + off);
        atomicMax(&smax[zv.x & (NUM_ZONES - 1)], fmap(xv.x));  // ds_max_u32
        atomicMax(&smax[zv.y & (NUM_ZONES - 1)], fmap(xv.y));
        atomicMax(&smax[zv.z & (NUM_ZONES - 1)], fmap(xv.z));
        atomicMax(&smax[zv.w & (NUM_ZONES - 1)], fmap(xv.w));
    }
    __syncthreads();

    const unsigned int v = smax[tid];            // tid == zone id (TPB==NUM_ZONES)
    if (v)                                       // skip empty zones: fewer L2 atomics
        atomicMax(&gmax[(size_t)blockIdx.y * NUM_ZONES + tid], v);
}

// ---------------------------------------------------------------------------
// Kernel 2: mask = (x attains its zone's max), as 0.0f / 1.0f, plus count.
// Seg-max table cached in LDS. Mask stored non-temporally (TH=NT) so the
// 64 MB streaming output does not evict x/z lines still resident in the
// 192 MB L2 from pass 1.
// ---------------------------------------------------------------------------
__global__ void ce_mask(const float* __restrict__ x,
                        const int*   __restrict__ z,
                        const unsigned int* __restrict__ gmax,
                        float* __restrict__ mask,
                        unsigned int* __restrict__ counter,
                        int n_per_batch) {
    __shared__ unsigned int smax[NUM_ZONES];
    __shared__ unsigned int scount;
    const int tid = threadIdx.x;
    smax[tid] = gmax[(size_t)blockIdx.y * NUM_ZONES + tid];
    if (tid == 0) scount = 0u;
    __syncthreads();

    const size_t bbase  = (size_t)blockIdx.y * (size_t)n_per_batch;
    const int    stride = (int)gridDim.x * TPB * VEC;
    unsigned int local = 0u;

    for (int off = ((int)blockIdx.x * TPB + tid) * VEC; off < n_per_batch; off += stride) {
        const f32x4 xv = *(const f32x4*)(x + bbase + off);
        const i32x4 zv = *(const i32x4*)(z + bbase + off);
        const unsigned int m0 = (fmap(xv.x) == smax[zv.x & (NUM_ZONES - 1)]) ? 1u : 0u;
        const unsigned int m1 = (fmap(xv.y) == smax[zv.y & (NUM_ZONES - 1)]) ? 1u : 0u;
        const unsigned int m2 = (fmap(xv.z) == smax[zv.z & (NUM_ZONES - 1)]) ? 1u : 0u;
        const unsigned int m3 = (fmap(xv.w) == smax[zv.w & (NUM_ZONES - 1)]) ? 1u : 0u;
        f32x4 o;
        o.x = (float)m0; o.y = (float)m1; o.z = (float)m2; o.w = (float)m3;
        __builtin_nontemporal_store(o, (f32x4*)(mask + bbase + off)); // global_store_b128 NT
        local += m0 + m1 + m2 + m3;
    }

    if (local) atomicAdd(&scount, local);        // ds_add_u32
    __syncthreads();
    if (tid == 0 && scount) atomicAdd(counter, scount);
}

// ---------------------------------------------------------------------------
// Kernel 3: n_centers (int sum in reference) -> float in d_out's tail slot.
// Max possible count = 64*262144 = 2^24 (ties all count), exact in f32.
// ---------------------------------------------------------------------------
__global__ void ce_finalize(const unsigned int* __restrict__ counter,
                            float* __restrict__ out_scalar) {
    if (threadIdx.x == 0 && blockIdx.x == 0)
        out_scalar[0] = (float)(*counter);
}

extern "C" void kernel_launch(void* const* d_in, const int* in_sizes, int n_in,
                              void* d_out, int out_size, void* d_ws, size_t ws_size,
                              hipStream_t stream) {
    const float* x = (const float*)d_in[0];   // [B,1,H,W] f32
    const int*   z = (const int*)d_in[1];     // [B,H,W]   i32

    const int total = in_sizes[0];            // B*H*W = 16,777,216
    const int B     = 64;                     // fixed by setup_inputs()
    const int N     = total / B;              // 512*512 = 262,144

    // Workspace layout: u32 gmax[B*NUM_ZONES] | u32 counter
    unsigned int* gmax    = (unsigned int*)d_ws;
    unsigned int* counter = gmax + (size_t)B * NUM_ZONES;
    const int ws_words    = B * NUM_ZONES + 1;

    float* mask       = (float*)d_out;        // B*N mask elements
    float* out_scalar = mask + (size_t)total; // n_centers slot

    // 0) zero seg-max table + counter (re-run every call: deterministic)
    ce_init_ws<<<(ws_words + TPB - 1) / TPB, TPB, 0, stream>>>(gmax, ws_words);

    // 1) segmented max: 32 blocks per batch (grid-stride x8) keeps global
    //    atomic traffic ~0.5M ops over 16K addresses; 2048 blocks total.
    dim3 g1(32, B);
    ce_seg_max<<<g1, TPB, 0, stream>>>(x, z, gmax, N);

    // 2) mask + count: 256 blocks per batch, loop runs exactly once
    //    (256 blocks * 256 thr * 4 = N). x/z mostly hit L2 from pass 1.
    dim3 g2(N / (TPB * VEC), B);
    ce_mask<<<g2, TPB, 0, stream>>>(x, z, gmax, mask, counter, N);

    // 3) scalar output
    ce_finalize<<<1, 32, 0, stream>>>(counter, out_scalar);
}